// CSGCL_12489764897131
// MI455X (gfx1250) — compile-verified
//
#include <hip/hip_runtime.h>

// ---------------------------------------------------------------------------
// 2-layer GCN for MI455X (gfx1250, wave32).
//   layer: H = A_in @ W ; out = relu( scatter(H) + H*dis^2 + b )
// GEMM: V_WMMA_F32_16X16X4_F32, 64 M-rows per block, each wave holds 4
// accumulator tiles so every B fragment feeds 4 back-to-back WMMAs.
// Edge scatter: global_atomic_add_f32 (agg buffers 51MB/26MB -> L2-resident).
// edge_index is jnp.int64 in the reference -> read as 64-bit ints.
// ---------------------------------------------------------------------------

typedef __attribute__((ext_vector_type(2))) float v2f;
typedef __attribute__((ext_vector_type(8))) float v8f;

#define MROWS 64   // M rows per block = 4 WMMA tiles in M per wave

// ---------------- init / degree / rsqrt ----------------

__global__ void gcn_fill1(float* __restrict__ p, int n) {
    int i = blockIdx.x * blockDim.x + threadIdx.x;
    if (i < n) p[i] = 1.0f;   // self-loop contribution to degree
}

__global__ void gcn_degree(const long long* __restrict__ ei, float* __restrict__ deg,
                           long long E) {
    long long e = (long long)blockIdx.x * blockDim.x + threadIdx.x;
    if (e < E) {
        long long dst = ei[E + e];
        atomicAdd(&deg[dst], 1.0f);
    }
}

__global__ void gcn_rsqrt(float* __restrict__ deg, int n) {
    int i = blockIdx.x * blockDim.x + threadIdx.x;
    if (i < n) deg[i] = rsqrtf(deg[i]);   // deg >= 1 -> finite
}

// ---------------- WMMA fp32 GEMM:  H[M,Nout] = A[M,K] @ W[K,Nout] ------------
// Block: MROWS(64) M-rows, (Nout/16) waves; wave w owns N-tile n0=16w and the
// 4 M-tiles -> 4 v8f accumulators. A staged in LDS (stride K+4: column reads
// spread over 16 banks). Per k-step: 1 B frag (2 coalesced global loads,
// L2-resident W) reused by 4 WMMAs fed by 4 ds_load_b64 A frags.
// A frag (16x4 f32): lanes 0-15 hold K={k0,k0+1}, lanes 16-31 K={k0+2,k0+3}.
// C/D (16x16 f32): VGPR r, lanes 0-15 -> M=r, lanes 16-31 -> M=r+8.

__global__ void gcn_gemm_wmma(const float* __restrict__ A, const float* __restrict__ W,
                              float* __restrict__ H, int M, int K, int Nout) {
    extern __shared__ float ldsA[];          // [MROWS][K+4]
    const int stride = K + 4;
    const int tid  = threadIdx.x;
    const int T    = blockDim.x;
    const int m0   = blockIdx.x * MROWS;

    // cooperative float4 load of the MROWSxK A tile (rows clamped at tail)
    const int vecs = (MROWS * K) >> 2;
    for (int i = tid; i < vecs; i += T) {
        int row = (4 * i) / K;
        int col = (4 * i) % K;
        int gr  = m0 + row; gr = (gr < M) ? gr : (M - 1);
        float4 v = *(const float4*)(A + (size_t)gr * K + col);
        *(float4*)(ldsA + row * stride + col) = v;
    }
    __syncthreads();

    const int lane = tid & 31;
    const int wave = tid >> 5;
    const int l16  = lane & 15;
    const int koff = (lane >> 4) << 1;       // 0 for lanes 0-15, 2 for 16-31
    const int n0   = wave << 4;

    v8f c[4];
#pragma unroll
    for (int mt = 0; mt < 4; ++mt) c[mt] = (v8f){0.f,0.f,0.f,0.f,0.f,0.f,0.f,0.f};

    for (int k0 = 0; k0 < K; k0 += 4) {
        const float* bp = W + (size_t)(k0 + koff) * Nout + (n0 + l16);
        v2f b; b.x = bp[0]; b.y = bp[Nout];  // coalesced; reused 4x below
#pragma unroll
        for (int mt = 0; mt < 4; ++mt) {
            const float* ap = ldsA + (mt * 16 + l16) * stride + (k0 + koff);
            v2f a; a.x = ap[0]; a.y = ap[1]; // ds_load_b64 (8B aligned)
            c[mt] = __builtin_amdgcn_wmma_f32_16x16x4_f32(
                        /*neg_a=*/false, a, /*neg_b=*/false, b,
                        /*c_mod=*/(short)0, c[mt],
                        /*reuse_a=*/false, /*reuse_b=*/false);
        }
    }

    const int hiM = (lane >> 4) << 3;        // +8 for lanes 16-31
#pragma unroll
    for (int mt = 0; mt < 4; ++mt) {
        const int mbase = m0 + mt * 16 + hiM;
#pragma unroll
        for (int r = 0; r < 8; ++r) {
            int rr = mbase + r;
            if (rr < M) H[(size_t)rr * Nout + (n0 + l16)] = c[mt][r];
        }
    }
}

// ---------------- edge scatter: agg[dst] += h[src] * dis[src]*dis[dst] -------
// F/4 lanes per edge; each lane: one global_load_b128 + 4 f32 atomics.
// (F=128 -> 1 edge/wave; F=64 -> 2 edges/wave.)

template <int F>
__global__ void gcn_scatter(const long long* __restrict__ ei,
                            const float* __restrict__ h,
                            const float* __restrict__ dis,
                            float* __restrict__ agg,
                            long long E) {
    constexpr int LPE = F / 4;               // lanes per edge
    long long gid = (long long)blockIdx.x * blockDim.x + threadIdx.x;
    long long e   = gid / LPE;
    int sub       = (int)(gid % LPE);        // lane's float4 slot within edge
    if (e >= E) return;
    long long src = ei[e];
    long long dst = ei[E + e];
    float norm = dis[src] * dis[dst];
    const float* hp = h   + (size_t)src * F + sub * 4;
    float*       ap = agg + (size_t)dst * F + sub * 4;
    float4 v = *(const float4*)hp;           // global_load_b128
    atomicAdd(ap + 0, v.x * norm);
    atomicAdd(ap + 1, v.y * norm);
    atomicAdd(ap + 2, v.z * norm);
    atomicAdd(ap + 3, v.w * norm);
}

// ---------------- combine: out = relu(agg + h*dis^2 + b), float4-wide --------

__global__ void gcn_combine(const float4* __restrict__ agg, const float4* __restrict__ h,
                            const float* __restrict__ dis, const float4* __restrict__ b,
                            float4* __restrict__ out, long long total4, int F4) {
    long long i = (long long)blockIdx.x * blockDim.x + threadIdx.x;
    if (i >= total4) return;
    long long node = i / F4;
    int f4 = (int)(i % F4);
    float d  = dis[node];
    float dd = d * d;
    float4 a  = agg[i];
    float4 hv = h[i];
    float4 bv = b[f4];
    float4 o;
    o.x = fmaxf(fmaf(hv.x, dd, a.x) + bv.x, 0.0f);
    o.y = fmaxf(fmaf(hv.y, dd, a.y) + bv.y, 0.0f);
    o.z = fmaxf(fmaf(hv.z, dd, a.z) + bv.z, 0.0f);
    o.w = fmaxf(fmaf(hv.w, dd, a.w) + bv.w, 0.0f);
    out[i] = o;
}

// ---------------- launch -----------------------------------------------------

extern "C" void kernel_launch(void* const* d_in, const int* in_sizes, int n_in,
                              void* d_out, int out_size, void* d_ws, size_t ws_size,
                              hipStream_t stream) {
    const float*      x  = (const float*)d_in[0];
    const long long*  ei = (const long long*)d_in[1];   // int64 per reference
    const float*      W1 = (const float*)d_in[2];
    const float*      b1 = (const float*)d_in[3];
    const float*      W2 = (const float*)d_in[4];
    const float*      b2 = (const float*)d_in[5];
    float*            out = (float*)d_out;

    const int IN  = 256, HID = 128, OUT = 64;
    const int N   = in_sizes[0] / IN;                   // 100000
    const long long E = (long long)in_sizes[1] / 2;     // 1600000

    // workspace partition (256B aligned); total ~154MB
    size_t off = 0;
    auto carve = [&](size_t bytes) -> float* {
        float* p = (float*)((char*)d_ws + off);
        off = (off + bytes + 255) & ~(size_t)255;
        return p;
    };
    float* dis  = carve((size_t)N * 4);          // degree, then rsqrt in place
    float* H1   = carve((size_t)N * HID * 4);
    float* agg1 = carve((size_t)N * HID * 4);    // becomes Y1 in place
    float* H2   = carve((size_t)N * OUT * 4);
    float* agg2 = carve((size_t)N * OUT * 4);

    // zero aggregation buffers every call (deterministic, graph-capture safe)
    hipMemsetAsync(agg1, 0, (size_t)N * HID * 4, stream);
    hipMemsetAsync(agg2, 0, (size_t)N * OUT * 4, stream);

    // degrees with self-loops -> dis = rsqrt(deg)
    gcn_fill1 <<<(N + 255) / 256, 256, 0, stream>>>(dis, N);
    gcn_degree<<<(int)((E + 255) / 256), 256, 0, stream>>>(ei, dis, E);
    gcn_rsqrt <<<(N + 255) / 256, 256, 0, stream>>>(dis, N);

    const int mtiles = (N + MROWS - 1) / MROWS;

    // ---- layer 1 ----
    gcn_gemm_wmma<<<mtiles, (HID / 16) * 32, MROWS * (IN + 4) * 4, stream>>>(
        x, W1, H1, N, IN, HID);
    {
        long long threads = E * (HID / 4);
        gcn_scatter<128><<<(int)((threads + 255) / 256), 256, 0, stream>>>(
            ei, H1, dis, agg1, E);
    }
    {
        long long total4 = (long long)N * HID / 4;
        gcn_combine<<<(int)((total4 + 255) / 256), 256, 0, stream>>>(
            (const float4*)agg1, (const float4*)H1, dis, (const float4*)b1,
            (float4*)agg1 /*in place -> Y1*/, total4, HID / 4);
    }

    // ---- layer 2 ----
    gcn_gemm_wmma<<<mtiles, (OUT / 16) * 32, MROWS * (HID + 4) * 4, stream>>>(
        agg1, W2, H2, N, HID, OUT);
    {
        long long threads = E * (OUT / 4);
        gcn_scatter<64><<<(int)((threads + 255) / 256), 256, 0, stream>>>(
            ei, H2, dis, agg2, E);
    }
    {
        long long total4 = (long long)N * OUT / 4;
        gcn_combine<<<(int)((total4 + 255) / 256), 256, 0, stream>>>(
            (const float4*)agg2, (const float4*)H2, dis, (const float4*)b2,
            (float4*)out, total4, OUT / 4);
    }
}